// CompressionHead_26010321945257
// MI455X (gfx1250) — compile-verified
//
#include <hip/hip_runtime.h>
#include <math.h>

typedef __attribute__((ext_vector_type(2))) float v2f;
typedef __attribute__((ext_vector_type(8))) float v8f;

#define BB 4
#define SS 4096
#define DD 2048
#define EPSN 1e-12f

#define ROW_SPLITS 16
#define CTILES (DD / 16)                 /* 128 column tiles of 16 dims   */
#define ROWS_PER_SPLIT (SS / ROW_SPLITS) /* 256 rows per wave             */

/* d_ws float layout:
 *   [0, BB*SS)              inv_norm per row
 *   [BB*SS, BB*SS+BB*DD)    colsum accumulator (t_{b,d})
 *   [BB*SS+BB*DD]           diag_sum accumulator
 */
#define WS_INV 0
#define WS_COL (BB * SS)
#define WS_DIAG (BB * SS + BB * DD)

__global__ void k_init(float* __restrict__ ws) {
    int i = blockIdx.x * blockDim.x + threadIdx.x;
    if (i < BB * DD + 1) ws[WS_COL + i] = 0.0f;
}

/* One wave (32 lanes) per row: sum of squares -> inv_norm + diag term. */
__global__ void k_norms(const float* __restrict__ h, float* __restrict__ ws) {
    float* inv_norm = ws + WS_INV;
    float* diag     = ws + WS_DIAG;
    const int wave = threadIdx.x >> 5;
    const int lane = threadIdx.x & 31;
    const int row  = blockIdx.x * 8 + wave;          /* [0, BB*SS) exactly */
    const float* p = h + (size_t)row * DD;

    float ss = 0.0f;
    for (int j = 0; j < DD; j += 128) {              /* 32 lanes * float4 */
        const float4 v = *(const float4*)(p + j + lane * 4);
        ss += v.x * v.x + v.y * v.y + v.z * v.z + v.w * v.w;
    }
    for (int off = 16; off > 0; off >>= 1)
        ss += __shfl_xor(ss, off, 32);

    if (lane == 0) {
        const float nrm = sqrtf(ss);
        const float inv = 1.0f / fmaxf(nrm, EPSN);
        inv_norm[row] = inv;
        const float d = nrm * inv;                   /* == sim_ii */
        atomicAdd(diag, d * d);
    }
}

/* Column sums of normalized rows via V_WMMA_F32_16X16X4_F32 with A = ones.
 * Each wave: one (batch, 16-dim tile, 256-row slice); 64 chained WMMAs. */
__global__ void k_colsum(const float* __restrict__ h, float* __restrict__ ws) {
    const float* inv_norm = ws + WS_INV;
    float* colsum         = ws + WS_COL;

    const int wave = threadIdx.x >> 5;
    const int lane = threadIdx.x & 31;
    const int gw   = blockIdx.x * 8 + wave;          /* [0, BB*CTILES*ROW_SPLITS) */
    const int b    = gw / (CTILES * ROW_SPLITS);
    const int rem  = gw % (CTILES * ROW_SPLITS);
    const int c    = rem / ROW_SPLITS;
    const int r0   = (rem % ROW_SPLITS) * ROWS_PER_SPLIT;

    const int n  = lane & 15;                        /* column within tile */
    const int kh = lane >> 4;                        /* K half select     */

    const float* base = h + ((size_t)b * SS) * DD + (size_t)(c * 16 + n);
    const float* inb  = inv_norm + b * SS;

    v2f a;  a.x = 1.0f; a.y = 1.0f;                  /* A = 16x4 ones */
    v8f acc = {};

    for (int r = r0; r < r0 + ROWS_PER_SPLIT; r += 4) {
        const int ra = r + kh;                       /* rows r..r+3 split  */
        const int rb = r + 2 + kh;                   /* across half-waves  */
        v2f bm;
        bm.x = base[(size_t)ra * DD] * inb[ra];
        bm.y = base[(size_t)rb * DD] * inb[rb];
        acc = __builtin_amdgcn_wmma_f32_16x16x4_f32(
            /*neg_a=*/false, a, /*neg_b=*/false, bm,
            /*c_mod=*/(short)0, acc, /*reuse_a=*/false, /*reuse_b=*/false);
    }

    /* A all-ones => every D row holds the column sums; take M=0 row. */
    if (lane < 16)
        atomicAdd(&colsum[b * DD + c * 16 + lane], acc[0]);
}

__global__ void k_final(const float* __restrict__ ws,
                        const float* __restrict__ alpha_p,
                        const float* __restrict__ beta_p,
                        float* __restrict__ out) {
    __shared__ float red[256];
    const float* colsum = ws + WS_COL;
    float s = 0.0f;
    for (int i = threadIdx.x; i < BB * DD; i += 256) {
        const float t = colsum[i];
        s += t * t;                                  /* Σ_b ‖Σ_i u_i‖²    */
    }
    red[threadIdx.x] = s;
    __syncthreads();
    for (int o = 128; o > 0; o >>= 1) {
        if ((int)threadIdx.x < o) red[threadIdx.x] += red[threadIdx.x + o];
        __syncthreads();
    }
    if (threadIdx.x == 0) {
        const float total = red[0];
        const float diag  = ws[WS_DIAG];
        const float denom = (float)BB * (float)SS * (float)(SS - 1);
        const float conc  = (total - diag) / denom;
        const float lam   = 1.0f / (1.0f + expf(-(*alpha_p) * (conc - (*beta_p))));
        out[0] = lam;
        out[1] = conc;
    }
}

extern "C" void kernel_launch(void* const* d_in, const int* in_sizes, int n_in,
                              void* d_out, int out_size, void* d_ws, size_t ws_size,
                              hipStream_t stream) {
    (void)in_sizes; (void)n_in; (void)out_size; (void)ws_size;
    const float* h     = (const float*)d_in[0];
    const float* alpha = (const float*)d_in[1];
    const float* beta  = (const float*)d_in[2];
    float* out = (float*)d_out;
    float* ws  = (float*)d_ws;

    k_init  <<<(BB * DD + 1 + 255) / 256, 256, 0, stream>>>(ws);
    k_norms <<<(BB * SS) / 8, 256, 0, stream>>>(h, ws);
    k_colsum<<<(BB * CTILES * ROW_SPLITS) / 8, 256, 0, stream>>>(h, ws);
    k_final <<<1, 256, 0, stream>>>(ws, alpha, beta, out);
}